// ListNetT1Loss_31482110280450
// MI455X (gfx1250) — compile-verified
//
#include <hip/hip_runtime.h>
#include <hip/hip_bf16.h>
#include <stdint.h>

// ---------------------------------------------------------------------------
// CDNA5 feature detection (compile-only loop: keep everything guarded)
// ---------------------------------------------------------------------------
#if __has_builtin(__builtin_amdgcn_global_load_async_to_lds_b32)
#define HAS_ASYNC_LDS 1
#else
#define HAS_ASYNC_LDS 0
#endif

#if __has_builtin(__builtin_amdgcn_wmma_f32_16x16x4_f32)
#define HAS_WMMA4 1
#else
#define HAS_WMMA4 0
#endif

typedef __attribute__((ext_vector_type(2))) float v2f;
typedef __attribute__((ext_vector_type(8))) float v8f;

__device__ __forceinline__ void async_copy_f32(float* lds_dst, const float* gsrc) {
#if HAS_ASYNC_LDS
  __builtin_amdgcn_global_load_async_to_lds_b32(
      (__attribute__((address_space(1))) int*)(uintptr_t)gsrc,
      (__attribute__((address_space(3))) int*)(uint32_t)(uintptr_t)lds_dst,
      /*offset=*/0, /*cpol=*/0);
#else
  *lds_dst = *gsrc;
#endif
}

__device__ __forceinline__ void async_wait_all() {
#if HAS_ASYNC_LDS
#if __has_builtin(__builtin_amdgcn_s_wait_asynccnt)
  __builtin_amdgcn_s_wait_asynccnt(0);
#else
  asm volatile("s_wait_asynccnt 0" ::: "memory");
#endif
#endif
}

__device__ __forceinline__ float wave_max32(float v) {
#pragma unroll
  for (int o = 16; o > 0; o >>= 1) v = fmaxf(v, __shfl_xor(v, o, 32));
  return v;
}
__device__ __forceinline__ float wave_sum32(float v) {
#pragma unroll
  for (int o = 16; o > 0; o >>= 1) v += __shfl_xor(v, o, 32);
  return v;
}

// ---------------------------------------------------------------------------
// Kernel 1: exclusive prefix sum of segment lengths (S <= 16384), one block.
// ---------------------------------------------------------------------------
__global__ __launch_bounds__(1024) void scan_offsets_kernel(const int* __restrict__ scope,
                                                            int* __restrict__ offsets,
                                                            int S) {
  __shared__ int part[1024];
  const int tid = threadIdx.x;
  const int base = tid * 16;
  int vals[16];
  int s = 0;
#pragma unroll
  for (int i = 0; i < 16; ++i) {
    int idx = base + i;
    vals[i] = (idx < S) ? scope[idx] : 0;
    s += vals[i];
  }
  part[tid] = s;
  __syncthreads();
  // Hillis-Steele inclusive scan over 1024 partials
  for (int o = 1; o < 1024; o <<= 1) {
    int v = (tid >= o) ? part[tid - o] : 0;
    __syncthreads();
    part[tid] += v;
    __syncthreads();
  }
  int run = (tid == 0) ? 0 : part[tid - 1];
#pragma unroll
  for (int i = 0; i < 16; ++i) {
    int idx = base + i;
    if (idx < S) offsets[idx] = run;
    run += vals[i];
  }
}

// ---------------------------------------------------------------------------
// Kernel 2: one 256-thread block per segment.
//   Stage (t, m) tile into LDS via async global->LDS (ASYNCcnt path), then:
//   pass A: tmax, mmax ; pass B: st, sm, sw ; loss = log(sm) - sw/st
//   Hot loops are specialized so the staged path uses pure ds_load (no FLAT).
//   Wave-level sum reduction of (st, sm) done with ONE v_wmma_f32_16x16x4_f32.
// ---------------------------------------------------------------------------
#define TILE 4096
#define NTHREADS 256

__global__ __launch_bounds__(NTHREADS) void seg_loss_kernel(const float* __restrict__ means,
                                                            const float* __restrict__ targets,
                                                            const int* __restrict__ scope,
                                                            const int* __restrict__ offsets,
                                                            float* __restrict__ losses) {
  __shared__ float lds_t[TILE];
  __shared__ float lds_m[TILE];
  __shared__ float red[24];   // 8 waves x 3 values
  __shared__ float bcast[2];

  const int seg = blockIdx.x;
  const int len = scope[seg];
  const int off = offsets[seg];
  const int tid = threadIdx.x;
  const int lane = tid & 31;
  const int wid = tid >> 5;

  const float* __restrict__ gt = targets + off;
  const float* __restrict__ gm = means + off;

  const bool use_lds = (len <= TILE);
  if (use_lds) {
    for (int i = tid; i < len; i += NTHREADS) {
      async_copy_f32(&lds_t[i], gt + i);
      async_copy_f32(&lds_m[i], gm + i);
    }
    async_wait_all();
  }
  __syncthreads();

  // -------- Pass A: per-segment maxima of t and m --------
  float tmax = -3.402823466e+38f, mmax = -3.402823466e+38f;
  if (use_lds) {
    for (int i = tid; i < len; i += NTHREADS) {       // ds_load path
      tmax = fmaxf(tmax, lds_t[i]);
      mmax = fmaxf(mmax, lds_m[i]);
    }
  } else {
    for (int i = tid; i < len; i += NTHREADS) {       // global_load path
      tmax = fmaxf(tmax, gt[i]);
      mmax = fmaxf(mmax, gm[i]);
    }
  }
  tmax = wave_max32(tmax);
  mmax = wave_max32(mmax);
  if (lane == 0) { red[wid] = tmax; red[8 + wid] = mmax; }
  __syncthreads();
  if (tid == 0) {
    float a = red[0], b = red[8];
#pragma unroll
    for (int w = 1; w < 8; ++w) { a = fmaxf(a, red[w]); b = fmaxf(b, red[8 + w]); }
    bcast[0] = a; bcast[1] = b;
  }
  __syncthreads();
  tmax = bcast[0];
  mmax = bcast[1];

  // -------- Pass B: three fused sums --------
  float st = 0.f, sm = 0.f, sw = 0.f;
  if (use_lds) {
    for (int i = tid; i < len; i += NTHREADS) {       // ds_load path
      float e  = __expf(lds_t[i] - tmax);
      float dm = lds_m[i] - mmax;
      st += e;
      sm += __expf(dm);
      sw += e * dm;
    }
  } else {
    for (int i = tid; i < len; i += NTHREADS) {       // global_load path (L2 hit)
      float e  = __expf(gt[i] - tmax);
      float dm = gm[i] - mmax;
      st += e;
      sm += __expf(dm);
      sw += e * dm;
    }
  }

  // -------- Wave reduction of (st, sm) via one WMMA; sw via shuffles --------
  float st_tot, sm_tot;
#if HAS_WMMA4
  {
    // A (16x4 f32): lane L vgpr0 = A[L%16][2*(L>=16)+0], vgpr1 = A[L%16][2*(L>=16)+1]
    v2f a; a.x = st; a.y = sm;
    // B (4x16): columns 0..7 select even K (st), columns 8..15 select odd K (sm).
    v2f b; b.x = ((lane & 15) < 8) ? 1.0f : 0.0f; b.y = 1.0f - b.x;
    v8f c = {};
    c = __builtin_amdgcn_wmma_f32_16x16x4_f32(false, a, false, b, (short)0, c, false, false);
    float r = c[0] + c[1] + c[2] + c[3] + c[4] + c[5] + c[6] + c[7];
    r += __shfl_xor(r, 16, 32);       // combine rows 0-7 with rows 8-15
    st_tot = __shfl(r, 0, 32);        // column 0  -> st total
    sm_tot = __shfl(r, 8, 32);        // column 8  -> sm total
  }
#else
  st_tot = wave_sum32(st);
  sm_tot = wave_sum32(sm);
#endif
  float sw_tot = wave_sum32(sw);

  if (lane == 0) { red[wid] = st_tot; red[8 + wid] = sm_tot; red[16 + wid] = sw_tot; }
  __syncthreads();
  if (tid == 0) {
    float A = 0.f, B = 0.f, C = 0.f;
#pragma unroll
    for (int w = 0; w < 8; ++w) { A += red[w]; B += red[8 + w]; C += red[16 + w]; }
    losses[seg] = __logf(B) - C / A;
  }
}

// ---------------------------------------------------------------------------
// Kernel 3: deterministic tree reduction of per-segment losses -> scalar.
// ---------------------------------------------------------------------------
__global__ __launch_bounds__(1024) void final_reduce_kernel(const float* __restrict__ losses,
                                                            float* __restrict__ out, int S) {
  __shared__ float part[1024];
  const int tid = threadIdx.x;
  float s = 0.f;
  for (int i = tid; i < S; i += 1024) s += losses[i];
  part[tid] = s;
  __syncthreads();
  for (int o = 512; o > 0; o >>= 1) {
    if (tid < o) part[tid] += part[tid + o];
    __syncthreads();
  }
  if (tid == 0) out[0] = part[0] / (float)S;
}

// ---------------------------------------------------------------------------
// Host entry
// ---------------------------------------------------------------------------
extern "C" void kernel_launch(void* const* d_in, const int* in_sizes, int n_in,
                              void* d_out, int out_size, void* d_ws, size_t ws_size,
                              hipStream_t stream) {
  const float* means   = (const float*)d_in[0];
  const int*   scope   = (const int*)d_in[1];
  const float* targets = (const float*)d_in[2];
  const int S = in_sizes[1];

  int*   offsets = (int*)d_ws;
  float* losses  = (float*)((char*)d_ws + (size_t)S * sizeof(int));

  scan_offsets_kernel<<<1, 1024, 0, stream>>>(scope, offsets, S);
  seg_loss_kernel<<<S, NTHREADS, 0, stream>>>(means, targets, scope, offsets, losses);
  final_reduce_kernel<<<1, 1024, 0, stream>>>(losses, (float*)d_out, S);
}